// Cross_Att_6640019440172
// MI455X (gfx1250) — compile-verified
//
#include <hip/hip_runtime.h>
#include <hip/hip_bf16.h>
#include <math.h>

#if __has_include(<hip/amd_detail/amd_gfx1250_TDM.h>)
#define TDM_6ARG 1   // amdgpu-toolchain (clang-23, therock headers): 6-arg builtin
#else
#define TDM_6ARG 0   // ROCm 7.2 (clang-22): 5-arg builtin
#endif

#define EDGES 262144
#define DIM   256
#define DOUTC 256
#define NQC   65536
#define ALPHA 0.2f
#define EPS_V 1e-12f

#define TILES_TOTAL   (EDGES / 16)            // 16384 edge tiles
#define TILES_PER_BLK 8
#define NBLK          (TILES_TOTAL / TILES_PER_BLK)  // 2048 persistent blocks

typedef __attribute__((ext_vector_type(2))) float        v2f;
typedef __attribute__((ext_vector_type(8))) float        v8f;
typedef __attribute__((ext_vector_type(4))) unsigned int v4u;
typedef __attribute__((ext_vector_type(4))) int          v4i;
typedef __attribute__((ext_vector_type(8))) int          v8i;

// ---- TDM: DMA one 16x256 f32 key tile into LDS, HW-padded to 260-float rows ----
__device__ __forceinline__ void tdm_load_key_tile(const float* key_embed, size_t e0,
                                                  unsigned lds_byte) {
    unsigned long long gaddr =
        (unsigned long long)(uintptr_t)key_embed + (unsigned long long)e0 * DIM * 4ull;
    // D# group0: count=1 | lds_addr | global_addr | type=2
    v4u g0 = { 1u, lds_byte, (unsigned)gaddr,
               (unsigned)(gaddr >> 32) | (2u << 30) };
    // D# group1: data_size=4B, pad_enable, pad_interval=256dw(code7), pad_amount=4dw(code3)
    //            tensor_dim0=256, tensor_dim1=EDGES(0x40000), tile=256x16, dim0_stride=256
    v8i g1 = { (int)((2u << 16) | (1u << 20) | (7u << 22) | (3u << 25)),
               (int)(256u << 16),          // tensor_dim0[15:0] in [31:16]
               0,                          // tensor_dim0[31:16]=0, tensor_dim1[15:0]=0
               (int)(4u | (256u << 16)),   // tensor_dim1[31:16]=4, tile_dim0=256
               16,                         // tile_dim1=16, tile_dim2=0
               256,                        // tensor_dim0_stride lo32
               0, 0 };
    v4i z4 = { 0, 0, 0, 0 };               // groups 2/3 unused (2D tile)
#if TDM_6ARG
    v8i z8 = { 0, 0, 0, 0, 0, 0, 0, 0 };
    __builtin_amdgcn_tensor_load_to_lds(g0, g1, z4, z4, z8, 0);
#else
    __builtin_amdgcn_tensor_load_to_lds(g0, g1, z4, z4, 0);
#endif
}

// w[j] = sum_o a2[o] * a[o, j]   (fold a_2 into a: kills the (E,512)x(512,256) GEMM)
__global__ void k_fold_a2(const float* __restrict__ a, const float* __restrict__ a2,
                          float* __restrict__ w) {
    int j = blockIdx.x * blockDim.x + threadIdx.x;
    if (j >= 2 * DIM) return;
    float s = 0.f;
    #pragma unroll 4
    for (int o = 0; o < DOUTC; ++o) s = fmaf(a2[o], a[(size_t)o * (2 * DIM) + j], s);
    w[j] = s;
}

__global__ void k_zero(float* __restrict__ hprime, float* __restrict__ rowsum) {
    size_t i = (size_t)blockIdx.x * blockDim.x + threadIdx.x;
    size_t stride = (size_t)gridDim.x * blockDim.x;
    for (size_t j = i; j < (size_t)NQC * DOUTC; j += stride) hprime[j] = 0.f;
    for (size_t j = i; j < (size_t)NQC; j += stride) rowsum[j] = 0.f;
}

// Persistent 256-thread blocks (8 wave32); TDM double-buffers key tiles into LDS.
__global__ __launch_bounds__(256) void k_edge_tile(
    const float* __restrict__ key_embed, const int* __restrict__ query_list,
    const float* __restrict__ query_embed, const float* __restrict__ trans,
    const float* __restrict__ w, float* __restrict__ hprime, float* __restrict__ rowsum)
{
    __shared__ float lds_key[2][16 * 260];  // double buffer, 260-float padded rows
    __shared__ float s_part[16];
    __shared__ float s_ee[16];
    __shared__ int   s_q[16];

    const int tid  = threadIdx.x;
    const int wave = tid >> 5;
    const int lane = tid & 31;
    const int half = lane >> 4;
    const int lm   = lane & 15;
    const int o0   = wave * 32;
    const int t0   = blockIdx.x * TILES_PER_BLK;

    const unsigned lds_b0 = (unsigned)(uintptr_t)&lds_key[0][0];
    const unsigned lds_b1 = (unsigned)(uintptr_t)&lds_key[1][0];

    if (wave == 0) tdm_load_key_tile(key_embed, (size_t)t0 * 16, lds_b0);

    for (int t = 0; t < TILES_PER_BLK; ++t) {
        const int buf = t & 1;
        const size_t e0 = (size_t)(t0 + t) * 16;
        const float* kt = lds_key[buf];

        if (wave == 0) {
            if (t + 1 < TILES_PER_BLK) {   // prefetch next tile into other buffer
                tdm_load_key_tile(key_embed, e0 + 16, buf ? lds_b0 : lds_b1);
                __builtin_amdgcn_s_wait_tensorcnt(1);  // current tile done, next in flight
            } else {
                __builtin_amdgcn_s_wait_tensorcnt(0);
            }
        }
        if (tid < 16) { s_part[tid] = 0.f; s_q[tid] = query_list[e0 + tid]; }
        __syncthreads();   // publish TDM-written LDS + s_q to all waves

        // ---- s_e = w_k . k_e + w_q . q_e  (16 chunks of 32 per edge) ----
        {
            int m = tid & 15;
            int c = tid >> 4;
            float p = 0.f;
            if (c < 8) {
                int d0 = c * 32;
                #pragma unroll 8
                for (int i = 0; i < 32; ++i)
                    p = fmaf(w[d0 + i], kt[m * 260 + d0 + i], p);
            } else {
                int d0 = (c - 8) * 32;
                const float* qe = query_embed + (e0 + m) * DIM + d0;
                const float* wq = w + DIM + d0;
                #pragma unroll 8
                for (int i = 0; i < 32; ++i) p = fmaf(wq[i], qe[i], p);
            }
            atomicAdd(&s_part[m], p);          // ds_add_f32
        }
        __syncthreads();
        if (tid < 16) {
            float sv = s_part[tid];
            float lr = sv > 0.f ? sv : ALPHA * sv;
            float e  = __expf(-lr);
            s_ee[tid] = e;
            atomicAdd(&rowsum[s_q[tid]], e);   // global_atomic_add_f32 (L2-resident)
        }
        __syncthreads();

        // ---- edge_assoc tile: C(16x256) = Ktile(16x256) @ trans^T, WMMA f32 16x16x4 ----
        const v2f* ap  = (const v2f*)&kt[lm * 260 + 2 * half];
        const v2f* b0p = (const v2f*)(trans + (size_t)(o0 + lm) * DIM + 2 * half);
        const v2f* b1p = (const v2f*)(trans + (size_t)(o0 + 16 + lm) * DIM + 2 * half);

        v8f acc0 = {}; v8f acc1 = {};
        #pragma unroll 4
        for (int k = 0; k < 64; ++k) {
            v2f af = ap[2 * k];
            v2f b0 = b0p[2 * k];
            v2f b1 = b1p[2 * k];
            acc0 = __builtin_amdgcn_wmma_f32_16x16x4_f32(
                false, af, false, b0, (short)0, acc0, false, false);
            acc1 = __builtin_amdgcn_wmma_f32_16x16x4_f32(
                false, af, false, b1, (short)0, acc1, false, false);
        }

        // ---- scale rows by edge_e, scatter-add into h_prime[q] ----
        #pragma unroll
        for (int v = 0; v < 8; ++v) {
            int m = v + 8 * half;              // C/D layout: VGPR v -> row v + 8*(lane/16)
            float e = s_ee[m];
            size_t rowbase = (size_t)s_q[m] * DOUTC;
            atomicAdd(&hprime[rowbase + o0 + lm],      acc0[v] * e);
            atomicAdd(&hprime[rowbase + o0 + 16 + lm], acc1[v] * e);
        }
        __syncthreads();   // protect LDS buffers (s_*, key buf) before next tile
    }
}

__global__ void k_finish(float* __restrict__ hprime, const float* __restrict__ rowsum) {
    size_t i = (size_t)blockIdx.x * blockDim.x + threadIdx.x;
    if (i >= (size_t)NQC * DOUTC) return;
    float r = rowsum[i / DOUTC];
    r = (r == 0.f) ? EPS_V : r;
    float x = hprime[i] / r;
    hprime[i] = x > 0.f ? x : (__expf(x) - 1.f);   // elu
}

extern "C" void kernel_launch(void* const* d_in, const int* in_sizes, int n_in,
                              void* d_out, int out_size, void* d_ws, size_t ws_size,
                              hipStream_t stream) {
    // setup_inputs order: key_list, key_embed, query_list, query_embed, a, a_2, trans
    const float* key_embed   = (const float*)d_in[1];
    const int*   query_list  = (const int*)  d_in[2];
    const float* query_embed = (const float*)d_in[3];
    const float* a           = (const float*)d_in[4];
    const float* a2          = (const float*)d_in[5];
    const float* trans       = (const float*)d_in[6];

    float* hprime = (float*)d_out;        // accumulate h_prime directly in d_out
    float* w      = (float*)d_ws;         // [0, 512): folded a^T a_2
    float* rowsum = w + 2 * DIM;          // [512, 512+NQ): e_rowsum

    k_fold_a2<<<2, 256, 0, stream>>>(a, a2, w);
    k_zero<<<1024, 256, 0, stream>>>(hprime, rowsum);
    k_edge_tile<<<NBLK, 256, 0, stream>>>(key_embed, query_list, query_embed,
                                          trans, w, hprime, rowsum);
    k_finish<<<((size_t)NQC * DOUTC + 255) / 256, 256, 0, stream>>>(hprime, rowsum);
}